// TorchGridSampleParse_91225105367329
// MI455X (gfx1250) — compile-verified
//
#include <hip/hip_runtime.h>
#include <hip/hip_bf16.h>

// Problem shape (fixed by the reference): N=1, C=8, H=256, W=512, D=128
#define HW_TOTAL (256 * 512)
#define D_DIM    128
#define C_DIM    8
#define CH_STRIDE ((size_t)HW_TOTAL * D_DIM)   // 16,777,216 elements = 64 MB per channel

// One thread per pixel. Weights computed once per pixel; 8 channels unrolled
// -> 16 outstanding gather loads per thread.
//
// Addressing is deliberately "uniform base pointer + 32-bit lane offset" so
// the compiler emits GVS-mode loads (saddr = SGPR64, vaddr = u32, scale_offset)
// instead of per-lane 64-bit address arithmetic. Per-channel bases are
// wave-uniform -> SALU only.
//
// All gathered cachelines and output lines are single-use -> non-temporal
// cache policy (gfx1250 TH_NT) to avoid polluting the 192 MB L2.
__global__ __launch_bounds__(256) void grid_sample_d_kernel(
    const float* __restrict__ cv,    // (C, HW, D) contiguous
    const float* __restrict__ flow,  // (HW)
    float* __restrict__ out)         // (C, HW)
{
    // Grid is launched exactly: 512 blocks * 256 threads == HW_TOTAL.
    const unsigned hw = blockIdx.x * 256u + threadIdx.x;

    // x = flow * (D-1)/D  (align_corners grid_sample mapping from the reference)
    const float x   = flow[hw] * (127.0f / 128.0f);
    const float x0f = floorf(x);
    const float t   = x - x0f;          // weight of upper tap

    const int x0 = (int)x0f;
    const int x1 = x0 + 1;

    // zeros-padding masks folded into the weights (computed once per pixel)
    const float w0 = (1.0f - t) * ((x0 >= 0 && x0 < D_DIM) ? 1.0f : 0.0f);
    const float w1 = t          * ((x1 >= 0 && x1 < D_DIM) ? 1.0f : 0.0f);

    const int i0 = min(max(x0, 0), D_DIM - 1);
    const int i1 = min(max(x1, 0), D_DIM - 1);

    // 32-bit element offsets within one channel plane (max ~16.8M < 2^31).
    const unsigned o0 = hw * (unsigned)D_DIM + (unsigned)i0;
    const unsigned o1 = o0 + (unsigned)(i1 - i0);

#pragma unroll
    for (int c = 0; c < C_DIM; ++c) {
        const float* p = cv + (size_t)c * CH_STRIDE;   // wave-uniform (SGPR base)
        const float v0 = __builtin_nontemporal_load(p + o0);
        const float v1 = __builtin_nontemporal_load(p + o1);
        const float r  = v0 * w0 + v1 * w1;
        __builtin_nontemporal_store(r, out + (size_t)c * HW_TOTAL + hw);
    }
}

extern "C" void kernel_launch(void* const* d_in, const int* in_sizes, int n_in,
                              void* d_out, int out_size, void* d_ws, size_t ws_size,
                              hipStream_t stream) {
    (void)in_sizes; (void)n_in; (void)d_ws; (void)ws_size; (void)out_size;

    const float* cost_volume = (const float*)d_in[0];  // (1, 8, 131072, 128) f32
    const float* flow_map    = (const float*)d_in[1];  // (1, 256, 512, 1)   f32
    float*       out         = (float*)d_out;          // (1, 8, 256, 512)   f32

    const int threads = 256;                 // 8 wave32 waves per block
    const int blocks  = HW_TOTAL / threads;  // 512 blocks, exact cover
    grid_sample_d_kernel<<<blocks, threads, 0, stream>>>(cost_volume, flow_map, out);
}